// Four_Pos_Fusion_Embedding_35029753266234
// MI455X (gfx1250) — compile-verified
//
#include <hip/hip_runtime.h>

#define HIDDEN  128
#define SEQ     512
#define BATCH   2
#define TABLE   1025   // 2*512+1 table rows
#define MAXPOS  512

typedef float v2f __attribute__((ext_vector_type(2)));
typedef float v4f __attribute__((ext_vector_type(4)));
typedef float v8f __attribute__((ext_vector_type(8)));

// ---------------------------------------------------------------------------
// Kernel A: projected table  P[t, h] = sum_k pe[t, k] * W[h, koff + k]
// One wave per 16x16 output tile, V_WMMA_F32_16X16X4_F32 over K = 128.
// ---------------------------------------------------------------------------
__global__ __launch_bounds__(32) void proj_gemm_kernel(
    const float* __restrict__ pe,   // [TABLE, HIDDEN]
    const float* __restrict__ W,    // [HIDDEN, 2*HIDDEN]
    float* __restrict__ P,          // [TABLE, HIDDEN]
    int koff)                       // 0 for ss-half of W, HIDDEN for ee-half
{
    const int lane = threadIdx.x & 31;
    const int half = lane >> 4;       // 0: lanes 0-15, 1: lanes 16-31
    const int l15  = lane & 15;
    const int n0   = blockIdx.x * 16; // output column tile (h)
    const int t0   = blockIdx.y * 16; // table-row tile (m)

    // A-row clamp (rows >= TABLE read row TABLE-1; discarded at store time).
    int am = t0 + l15;
    if (am > TABLE - 1) am = TABLE - 1;
    const float* arow = pe + (size_t)am * HIDDEN + half * 2;
    // B: B[k][n] = W[n0+n][koff+k]; VGPR0 holds rows K,K+2 across lane halves.
    const float* brow = W + (size_t)(n0 + l15) * (2 * HIDDEN) + koff + half * 2;

    v8f c = {};
    for (int k0 = 0; k0 < HIDDEN; k0 += 4) {
        v2f a;  a.x = arow[k0];  a.y = arow[k0 + 1];
        v2f b;  b.x = brow[k0];  b.y = brow[k0 + 1];
        // (neg_a, A, neg_b, B, c_mod, C, reuse_a, reuse_b)
        c = __builtin_amdgcn_wmma_f32_16x16x4_f32(
                false, a, false, b, (short)0, c, false, false);
    }

    // D layout: VGPR r -> row M = t0 + r + half*8, col N = n0 + l15
#pragma unroll
    for (int r = 0; r < 8; ++r) {
        const int t = t0 + r + half * 8;
        if (t < TABLE)
            P[(size_t)t * HIDDEN + n0 + l15] = c[r];
    }
}

// ---------------------------------------------------------------------------
// Kernel B: out[b,i,j,:] = relu(P_ss[idx_ss] + P_ee[idx_ee] + bias)
// One wave per (b,i,j) row; 4 floats/lane; non-temporal b128 stores
// (256 MiB streaming output, zero reuse -> keep it out of L2).
// ---------------------------------------------------------------------------
__global__ __launch_bounds__(256) void fuse_kernel(
    const int*   __restrict__ pos_s,  // [BATCH, SEQ]
    const int*   __restrict__ pos_e,  // [BATCH, SEQ]
    const float* __restrict__ Pss,    // [TABLE, HIDDEN]
    const float* __restrict__ Pee,    // [TABLE, HIDDEN]
    const float* __restrict__ bias,   // [HIDDEN]
    float*       __restrict__ out)    // [BATCH, SEQ, SEQ, HIDDEN]
{
    const int lane = threadIdx.x & 31;
    const int wave = threadIdx.x >> 5;
    const long long r = (long long)blockIdx.x * 8 + wave;  // flat (b,i,j)

    const int j    = (int)(r & (SEQ - 1));
    const int bi   = (int)(r >> 9);       // b*SEQ + i   (SEQ == 512)
    const int b    = bi >> 9;
    const int i    = bi & (SEQ - 1);
    const int base = b * SEQ;

    const int iss = pos_s[base + i] - pos_s[base + j] + MAXPOS;
    const int iee = pos_e[base + i] - pos_e[base + j] + MAXPOS;

    const int h0 = lane * 4;
    const v4f s  = *(const v4f*)(Pss + (size_t)iss * HIDDEN + h0);
    const v4f e  = *(const v4f*)(Pee + (size_t)iee * HIDDEN + h0);
    const v4f bb = *(const v4f*)(bias + h0);

    v4f o;
    o.x = fmaxf(s.x + e.x + bb.x, 0.0f);
    o.y = fmaxf(s.y + e.y + bb.y, 0.0f);
    o.z = fmaxf(s.z + e.z + bb.z, 0.0f);
    o.w = fmaxf(s.w + e.w + bb.w, 0.0f);

    __builtin_nontemporal_store(o, (v4f*)(out + (size_t)r * HIDDEN + h0));
}

// ---------------------------------------------------------------------------
extern "C" void kernel_launch(void* const* d_in, const int* in_sizes, int n_in,
                              void* d_out, int out_size, void* d_ws, size_t ws_size,
                              hipStream_t stream) {
    (void)in_sizes; (void)n_in; (void)out_size; (void)ws_size;

    const int*   pos_s = (const int*)d_in[0];
    const int*   pos_e = (const int*)d_in[1];
    const float* pe_ss = (const float*)d_in[2];
    // d_in[3] (pe_se) and d_in[4] (pe_es) are dead in the reference.
    const float* pe_ee = (const float*)d_in[5];
    const float* W     = (const float*)d_in[6];
    const float* bias  = (const float*)d_in[7];
    float*       out   = (float*)d_out;

    float* Pss = (float*)d_ws;                       // [TABLE, HIDDEN]
    float* Pee = Pss + (size_t)TABLE * HIDDEN;       // [TABLE, HIDDEN]  (~1 MiB total)

    dim3 gA(HIDDEN / 16, (TABLE + 15) / 16);         // 8 x 65 tiles, 1 wave each
    proj_gemm_kernel<<<gA, 32, 0, stream>>>(pe_ss, W, Pss, 0);
    proj_gemm_kernel<<<gA, 32, 0, stream>>>(pe_ee, W, Pee, HIDDEN);

    const long long rows = (long long)BATCH * SEQ * SEQ;   // 524288
    fuse_kernel<<<(unsigned)(rows / 8), 256, 0, stream>>>(
        pos_s, pos_e, Pss, Pee, bias, out);
}